// PointNetSetAbstractionMSG_32409823215993
// MI455X (gfx1250) — compile-verified
//
#include <hip/hip_runtime.h>

typedef float v2f __attribute__((ext_vector_type(2)));
typedef float v4f __attribute__((ext_vector_type(4)));
typedef float v8f __attribute__((ext_vector_type(8)));

#define NPTS 4096
#define NFPS 1024
#define BATCH 8
#define OUTCH 320

// ---------------------------------------------------------------------------
// Furthest point sampling: one block per batch. min-distance array in LDS,
// block argmax with lowest-index tie-break (jnp.argmax first-max semantics).
// Writes new_xyz (gathered centers) straight into d_out.
// ---------------------------------------------------------------------------
__global__ __launch_bounds__(256)
void fps_kernel(const float* __restrict__ xyz, float* __restrict__ new_xyz)
{
    const int b   = blockIdx.x;
    const int tid = threadIdx.x;

    __shared__ float md[NPTS];
    __shared__ float rv[256];
    __shared__ int   ri[256];
    __shared__ int   s_last;

    const float* px = xyz + (size_t)b * 3 * NPTS;
    const float* py = px + NPTS;
    const float* pz = py + NPTS;

    for (int i = tid; i < NPTS; i += 256) md[i] = 1.0e10f;
    if (tid == 0) s_last = 0;
    __syncthreads();

    for (int it = 0; it < NFPS; ++it) {
        const int last = s_last;
        if (tid == 0) {
            new_xyz[((size_t)b * 3 + 0) * NFPS + it] = px[last];
            new_xyz[((size_t)b * 3 + 1) * NFPS + it] = py[last];
            new_xyz[((size_t)b * 3 + 2) * NFPS + it] = pz[last];
        }
        const float cx = px[last], cy = py[last], cz = pz[last];

        float bestv = -1.0f;
        int   besti = 0;
        for (int i = tid; i < NPTS; i += 256) {
            const float dx = px[i] - cx;
            const float dy = py[i] - cy;
            const float dz = pz[i] - cz;
            const float d  = dx * dx + dy * dy + dz * dz;
            float m = md[i];
            m = fminf(m, d);
            md[i] = m;
            if (m > bestv) { bestv = m; besti = i; }   // strict > keeps lowest index
        }
        rv[tid] = bestv;
        ri[tid] = besti;
        __syncthreads();
        for (int s = 128; s > 0; s >>= 1) {
            if (tid < s) {
                const float ov = rv[tid + s];
                const int   oi = ri[tid + s];
                if (ov > rv[tid] || (ov == rv[tid] && oi < ri[tid])) {
                    rv[tid] = ov; ri[tid] = oi;
                }
            }
            __syncthreads();
        }
        if (tid == 0) s_last = ri[0];
        __syncthreads();
    }
}

// ---------------------------------------------------------------------------
// Ball query: one thread per (batch, center). Scan points in ascending index
// order, take the first K hits within radius, pad with the first hit.
// ---------------------------------------------------------------------------
template <int K>
__global__ __launch_bounds__(256)
void ball_query_kernel(const float* __restrict__ xyz,
                       const float* __restrict__ new_xyz,
                       int* __restrict__ out_idx, float r2)
{
    const int g = blockIdx.x * 256 + threadIdx.x;   // b*NFPS + s
    if (g >= BATCH * NFPS) return;
    const int b = g / NFPS;
    const int s = g % NFPS;

    const float cx = new_xyz[((size_t)b * 3 + 0) * NFPS + s];
    const float cy = new_xyz[((size_t)b * 3 + 1) * NFPS + s];
    const float cz = new_xyz[((size_t)b * 3 + 2) * NFPS + s];

    const float* px = xyz + (size_t)b * 3 * NPTS;
    const float* py = px + NPTS;
    const float* pz = py + NPTS;

    int* dst  = out_idx + (size_t)g * K;
    int cnt   = 0;
    int first = NPTS - 1;
    for (int n = 0; n < NPTS && cnt < K; ++n) {
        const float dx = px[n] - cx;
        const float dy = py[n] - cy;
        const float dz = pz[n] - cz;
        const float d2 = dx * dx + dy * dy + dz * dz;
        if (d2 <= r2) {
            if (cnt == 0) first = n;
            dst[cnt++] = n;
        }
    }
    for (; cnt < K; ++cnt) dst[cnt] = first;
}

// ---------------------------------------------------------------------------
// One MLP layer: sout = relu(W * sin + bias) via V_WMMA_F32_16X16X4_F32.
// Activations live in LDS **column-major**: S[col][chan], stride SIN/SOUT
// floats (multiple of 4).  Per the WMMA VGPR layouts:
//   A 16x4 (c_out x k): lanes0-15 supply k = kb+0/+1, lanes16-31 k = kb+2/+3
//     -> one 8B load of W[row][k0..k0+1]
//   B 4x16 (k x col):   same k split per lane half, col = lane&15
//     -> one ds_load_b64 of sin[col][k0..k0+1]
//   C 16x16: VGPR r -> rows m0+r (lanes0-15) / m0+8+r (lanes16-31)
//     -> contiguous 8-float column run -> two ds_store_b128
// For CO in {32,64,128} each wave keeps one fixed M-tile and NTW N-tiles,
// reusing the A fragment NTW times per k-step.
// ---------------------------------------------------------------------------
template <int CI, int CO, int SIN, int SOUT>
__device__ inline void mlp_layer_wmma(const float* __restrict__ W,
                                      const float* __restrict__ bias,
                                      const float* sin_, float* sout,
                                      int lane, int wave)
{
    constexpr int MT = CO / 16;
    constexpr int NT = 4;                 // 64 cols
    const int mrow = lane & 15;
    const int hi   = lane >> 4;           // lane half
    const int koff = hi << 1;             // 0 or 2

    if constexpr ((8 % MT) == 0) {
        constexpr int WPM = 8 / MT;       // waves per M-tile
        constexpr int NTW = NT / WPM;     // N-tiles per wave (1,2,4)
        const int m0 = (wave % MT) * 16;
        const int n0 = (wave / MT) * NTW * 16;

        v8f acc[NTW];
#pragma unroll
        for (int j = 0; j < NTW; ++j) acc[j] = {};

#pragma unroll
        for (int kb = 0; kb < CI; kb += 4) {
            const int k0 = kb + koff;
            const v2f a = *(const v2f*)(W + (m0 + mrow) * CI + k0);
#pragma unroll
            for (int j = 0; j < NTW; ++j) {
                const int col = n0 + j * 16 + mrow;
                const v2f bm = *(const v2f*)(sin_ + col * SIN + k0);
                acc[j] = __builtin_amdgcn_wmma_f32_16x16x4_f32(
                    false, a, false, bm, (short)0, acc[j], false, false);
            }
        }

        const int rowbase = m0 + (hi << 3);
        const v4f blo = *(const v4f*)(bias + rowbase);
        const v4f bhi = *(const v4f*)(bias + rowbase + 4);
#pragma unroll
        for (int j = 0; j < NTW; ++j) {
            const int col = n0 + j * 16 + mrow;
            v4f lo, hh;
#pragma unroll
            for (int r = 0; r < 4; ++r) {
                lo[r] = fmaxf(acc[j][r]     + blo[r], 0.0f);
                hh[r] = fmaxf(acc[j][r + 4] + bhi[r], 0.0f);
            }
            float* dst = sout + col * SOUT + rowbase;
            *(v4f*)(dst)     = lo;
            *(v4f*)(dst + 4) = hh;
        }
    } else {
        // generic per-tile loop (CO = 96 -> 24 tiles / 8 waves)
        for (int t = wave; t < MT * NT; t += 8) {
            const int m0 = (t / NT) * 16;
            const int n0 = (t % NT) * 16;
            const int col = n0 + mrow;
            v8f acc = {};
#pragma unroll
            for (int kb = 0; kb < CI; kb += 4) {
                const int k0 = kb + koff;
                const v2f a  = *(const v2f*)(W + (m0 + mrow) * CI + k0);
                const v2f bm = *(const v2f*)(sin_ + col * SIN + k0);
                acc = __builtin_amdgcn_wmma_f32_16x16x4_f32(
                    false, a, false, bm, (short)0, acc, false, false);
            }
            const int rowbase = m0 + (hi << 3);
            const v4f blo = *(const v4f*)(bias + rowbase);
            const v4f bhi = *(const v4f*)(bias + rowbase + 4);
            v4f lo, hh;
#pragma unroll
            for (int r = 0; r < 4; ++r) {
                lo[r] = fmaxf(acc[r]     + blo[r], 0.0f);
                hh[r] = fmaxf(acc[r + 4] + bhi[r], 0.0f);
            }
            float* dst = sout + col * SOUT + rowbase;
            *(v4f*)(dst)     = lo;
            *(v4f*)(dst + 4) = hh;
        }
    }
}

// ---------------------------------------------------------------------------
// Fused group + 3-layer MLP + max-over-K. One workgroup handles CENTERS
// centers (CENTERS*K columns), processed in 64-column chunks through LDS.
// ---------------------------------------------------------------------------
template <int K, int C1, int C2, int C3, int CH_OFF>
__global__ __launch_bounds__(256)
void group_mlp_kernel(const float* __restrict__ xyz,
                      const float* __restrict__ new_xyz,
                      const int* __restrict__ idx,
                      const float* __restrict__ w1, const float* __restrict__ b1,
                      const float* __restrict__ w2, const float* __restrict__ b2,
                      const float* __restrict__ w3, const float* __restrict__ b3,
                      float* __restrict__ out)
{
    constexpr int COLS    = 64;
    constexpr int CENTERS = (K >= COLS) ? 1 : (COLS / K);
    constexpr int CHUNKS  = (K > COLS) ? (K / COLS) : 1;
    constexpr int CA      = (C1 > C3) ? C1 : C3;
    constexpr int SA      = CA + 4;       // strides (floats, multiple of 4)
    constexpr int SB      = C2 + 4;

    __shared__ float s0[COLS * 4];        // padded input, col-major [col][4]
    __shared__ float sA[COLS * SA];       // layer1 out / layer3 out
    __shared__ float sB[COLS * SB];       // layer2 out
    __shared__ float w1p[C1 * 4];         // W1 padded 3 -> 4 (col 3 = 0)
    __shared__ float smax[C3 * CENTERS];  // running max over K

    const int tid  = threadIdx.x;
    const int lane = tid & 31;
    const int wave = tid >> 5;

    const int wg = blockIdx.x;                    // B * (NFPS / CENTERS) blocks
    const int b  = wg / (NFPS / CENTERS);
    const int s0c = (wg % (NFPS / CENTERS)) * CENTERS;

    for (int p = tid; p < C3 * CENTERS; p += 256) smax[p] = -1.0e30f;
    for (int p = tid; p < C1 * 4; p += 256) {
        const int c = p >> 2, k = p & 3;
        w1p[p] = (k < 3) ? w1[c * 3 + k] : 0.0f;
    }

    const float* px = xyz + (size_t)b * 3 * NPTS;

    for (int ch = 0; ch < CHUNKS; ++ch) {
        __syncthreads();
        // ---- gather relative coordinates (col-major float4) ----
        for (int j = tid; j < COLS; j += 256) {
            const int colg = ch * COLS + j;
            const int sl = colg / K;
            const int k  = colg % K;
            const int s  = s0c + sl;
            const int n  = idx[((size_t)(b * NFPS + s)) * K + k];
            v4f g;
            g[0] = px[n]            - new_xyz[((size_t)b * 3 + 0) * NFPS + s];
            g[1] = px[NPTS + n]     - new_xyz[((size_t)b * 3 + 1) * NFPS + s];
            g[2] = px[2 * NPTS + n] - new_xyz[((size_t)b * 3 + 2) * NFPS + s];
            g[3] = 0.0f;
            *(v4f*)(s0 + j * 4) = g;
        }
        __syncthreads();

        mlp_layer_wmma<4,  C1, 4,  SA>(w1p, b1, s0, sA, lane, wave);
        __syncthreads();
        mlp_layer_wmma<C1, C2, SA, SB>(w2,  b2, sA, sB, lane, wave);
        __syncthreads();
        mlp_layer_wmma<C2, C3, SB, SA>(w3,  b3, sB, sA, lane, wave);
        __syncthreads();

        // ---- running max over K within this chunk (col-major reads) ----
        for (int p = tid; p < C3 * CENTERS; p += 256) {
            const int c  = p / CENTERS;
            const int sl = p % CENTERS;
            int lo = sl * K - ch * COLS;     if (lo < 0)    lo = 0;
            int hh = sl * K + K - ch * COLS; if (hh > COLS) hh = COLS;
            float m = smax[p];
            for (int j = lo; j < hh; ++j) m = fmaxf(m, sA[j * SA + c]);
            smax[p] = m;
        }
    }
    __syncthreads();

    for (int p = tid; p < C3 * CENTERS; p += 256) {
        const int c  = p / CENTERS;
        const int sl = p % CENTERS;
        out[((size_t)(b * OUTCH) + CH_OFF + c) * NFPS + (s0c + sl)] = smax[p];
    }
}

// ---------------------------------------------------------------------------
// Launch. Inputs: d_in[0]=xyz [8,3,4096], d_in[1]=points (unused),
// then 9 (w,b) pairs: scale0 L1..L3, scale1 L1..L3, scale2 L1..L3.
// d_out = new_xyz (8*3*1024) ++ features (8*320*1024).
// ---------------------------------------------------------------------------
extern "C" void kernel_launch(void* const* d_in, const int* in_sizes, int n_in,
                              void* d_out, int out_size, void* d_ws, size_t ws_size,
                              hipStream_t stream)
{
    (void)in_sizes; (void)n_in; (void)out_size; (void)ws_size;

    const float* xyz = (const float*)d_in[0];
    const float* w[9];
    const float* bb[9];
    for (int i = 0; i < 9; ++i) {
        w[i]  = (const float*)d_in[2 + 2 * i];
        bb[i] = (const float*)d_in[3 + 2 * i];
    }

    float* out_newxyz = (float*)d_out;                               // 24576 floats
    float* out_feat   = (float*)d_out + (size_t)BATCH * 3 * NFPS;    // 8*320*1024

    char* ws   = (char*)d_ws;
    int*  idx0 = (int*)ws;                                           // 8*1024*16
    int*  idx1 = (int*)(ws + (size_t)BATCH * NFPS * 16 * 4);         // 8*1024*32
    int*  idx2 = (int*)(ws + (size_t)BATCH * NFPS * (16 + 32) * 4);  // 8*1024*128

    fps_kernel<<<BATCH, 256, 0, stream>>>(xyz, out_newxyz);

    const int bq_blocks = (BATCH * NFPS + 255) / 256;   // 32
    ball_query_kernel<16 ><<<bq_blocks, 256, 0, stream>>>(xyz, out_newxyz, idx0, 0.1f * 0.1f);
    ball_query_kernel<32 ><<<bq_blocks, 256, 0, stream>>>(xyz, out_newxyz, idx1, 0.2f * 0.2f);
    ball_query_kernel<128><<<bq_blocks, 256, 0, stream>>>(xyz, out_newxyz, idx2, 0.4f * 0.4f);

    // scale 0: K=16, MLP 3->32->32->64, channels [0,64)
    group_mlp_kernel<16, 32, 32, 64, 0>
        <<<BATCH * (NFPS / 4), 256, 0, stream>>>(xyz, out_newxyz, idx0,
            w[0], bb[0], w[1], bb[1], w[2], bb[2], out_feat);
    // scale 1: K=32, MLP 3->64->64->128, channels [64,192)
    group_mlp_kernel<32, 64, 64, 128, 64>
        <<<BATCH * (NFPS / 2), 256, 0, stream>>>(xyz, out_newxyz, idx1,
            w[3], bb[3], w[4], bb[4], w[5], bb[5], out_feat);
    // scale 2: K=128, MLP 3->64->96->128, channels [192,320)
    group_mlp_kernel<128, 64, 96, 128, 192>
        <<<BATCH * NFPS, 256, 0, stream>>>(xyz, out_newxyz, idx2,
            w[6], bb[6], w[7], bb[7], w[8], bb[8], out_feat);
}